// Upsample_Reshape_Unit_15290083573883
// MI455X (gfx1250) — compile-verified
//
#include <hip/hip_runtime.h>
#include <stdint.h>

#define BATCH 16
#define CCH   256
#define NPTS  65536
#define LOUT  8192
#define TJ    64          // output positions per feature tile
#define PITCH 260         // floats per LDS row: 256 + 4 pad (keeps 16B align)

// ---------------------------------------------------------------------------
// Kernel 1: per-batch offsets/counts via binary search (batch ids are sorted).
// ---------------------------------------------------------------------------
__global__ void seg_bounds_kernel(const int* __restrict__ batch,
                                  int* __restrict__ offs,
                                  int* __restrict__ cnts) {
    __shared__ int bounds[BATCH + 1];
    int t = threadIdx.x;
    if (t <= BATCH) {
        int lo = 0, hi = NPTS;                  // lower_bound(batch, t)
        while (lo < hi) {
            int mid = (lo + hi) >> 1;
            if (batch[mid] < t) lo = mid + 1; else hi = mid;
        }
        bounds[t] = lo;
    }
    __syncthreads();
    if (t < BATCH) {
        offs[t] = bounds[t];
        cnts[t] = bounds[t + 1] - bounds[t];
    }
}

// ---------------------------------------------------------------------------
// Kernel 2: out_point [B,3,L]  (1.5 MB — trivial, coalesced stores along j)
// ---------------------------------------------------------------------------
__global__ __launch_bounds__(256) void upsample_points_kernel(
        const float* __restrict__ px,
        const int* __restrict__ offs,
        const int* __restrict__ cnts,
        float* __restrict__ outp) {
    int g = blockIdx.x * 256 + threadIdx.x;     // [0, B*L)
    int b = g >> 13;                             // L = 8192
    int j = g & (LOUT - 1);
    unsigned src = (unsigned)offs[b] + ((unsigned)j * (unsigned)cnts[b]) / LOUT;
    size_t sbase = (size_t)src * 3;
    float x = px[sbase + 0];
    float y = px[sbase + 1];
    float z = px[sbase + 2];
    size_t obase = (size_t)b * 3 * LOUT + j;
    outp[obase + 0 * LOUT] = x;
    outp[obase + 1 * LOUT] = y;
    outp[obase + 2 * LOUT] = z;
}

// ---------------------------------------------------------------------------
// Kernel 3: out_feature [B,C,L] gather+transpose, staged through LDS with the
// CDNA5 async global->LDS data mover (GLOBAL_LOAD_ASYNC_TO_LDS_B128 /
// ASYNCcnt / s_wait_asynccnt). One block = one (b, 64-j) tile x 256 channels.
// ---------------------------------------------------------------------------
__global__ __launch_bounds__(256) void upsample_features_kernel(
        const float* __restrict__ feat,
        const int* __restrict__ offs,
        const int* __restrict__ cnts,
        float* __restrict__ outf) {
    __shared__ float tile[TJ * PITCH];           // 66,560 B

    const int tid = threadIdx.x;
    const int tilesPerB = LOUT / TJ;             // 128
    const int b  = blockIdx.x / tilesPerB;
    const int j0 = (blockIdx.x % tilesPerB) * TJ;

    const unsigned offb = (unsigned)offs[b];
    const unsigned cntb = (unsigned)cnts[b];

    const uint32_t lds_base = (uint32_t)(uintptr_t)(&tile[0]);
    const uint64_t gbase    = (uint64_t)(uintptr_t)(feat);

    // ---- Load phase: 64 rows x 1KB, 16B per lane per async op.
    // Lane mapping: e = i*256+tid -> row jj = e>>6, float4 chunk cv = e&63.
    // A wave covers 512 contiguous bytes of one gathered feature row.
#pragma unroll
    for (int i = 0; i < 16; ++i) {
        int e  = i * 256 + tid;
        int jj = e >> 6;
        int cv = e & 63;
        unsigned src = offb + ((unsigned)(j0 + jj) * cntb) / LOUT;
        uint64_t gaddr = gbase + (uint64_t)src * (CCH * 4) + (uint64_t)(cv * 16);
        uint32_t laddr = lds_base + (uint32_t)(jj * (PITCH * 4) + cv * 16);
        asm volatile("global_load_async_to_lds_b128 %0, %1, off"
                     :: "v"(laddr), "v"(gaddr)
                     : "memory");
    }
    // Wait for this wave's async transfers, then make LDS visible block-wide.
    asm volatile("s_wait_asynccnt 0" ::: "memory");
    __syncthreads();

    // ---- Store phase: transpose out of LDS, float4 along j.
    // e = i*256+tid -> channel c = e>>4, j-quad g4 = e&15.
    // 16 consecutive lanes write 256 contiguous bytes of out[b,c,:].
#pragma unroll
    for (int i = 0; i < 16; ++i) {
        int e  = i * 256 + tid;
        int c  = e >> 4;
        int jj = (e & 15) * 4;
        float4 v;
        v.x = tile[(jj + 0) * PITCH + c];
        v.y = tile[(jj + 1) * PITCH + c];
        v.z = tile[(jj + 2) * PITCH + c];
        v.w = tile[(jj + 3) * PITCH + c];
        size_t oidx = ((size_t)b * CCH + c) * LOUT + (size_t)(j0 + jj);
        *reinterpret_cast<float4*>(outf + oidx) = v;
    }
}

// ---------------------------------------------------------------------------
extern "C" void kernel_launch(void* const* d_in, const int* in_sizes, int n_in,
                              void* d_out, int out_size, void* d_ws, size_t ws_size,
                              hipStream_t stream) {
    (void)in_sizes; (void)n_in; (void)out_size; (void)ws_size;

    const float* points_x       = (const float*)d_in[0];   // [N,3] f32
    const float* point_features = (const float*)d_in[1];   // [N,C] f32
    const int*   batch          = (const int*)d_in[2];     // [N] i32 sorted

    int* offs = (int*)d_ws;            // [B]
    int* cnts = offs + BATCH;          // [B]

    float* out_point   = (float*)d_out;                          // [B,3,L]
    float* out_feature = out_point + (size_t)BATCH * 3 * LOUT;   // [B,C,L]

    seg_bounds_kernel<<<1, 32, 0, stream>>>(batch, offs, cnts);

    upsample_points_kernel<<<(BATCH * LOUT) / 256, 256, 0, stream>>>(
        points_x, offs, cnts, out_point);

    upsample_features_kernel<<<BATCH * (LOUT / TJ), 256, 0, stream>>>(
        point_features, offs, cnts, out_feature);
}